// PoseEstimationHead_12068858101990
// MI455X (gfx1250) — compile-verified
//
#include <hip/hip_runtime.h>
#include <stdint.h>

typedef __attribute__((ext_vector_type(2))) float v2f;
typedef __attribute__((ext_vector_type(8))) float v8f;

#define BATCH   8
#define NPTS    1024
#define T_HYP   1000
#define HT      63            // hypothesis tiles of 16
#define T_PAD   (HT * 16)     // 1008
#define THRESH2 0.25f         // (0.5)^2

__device__ __forceinline__ uint32_t hash32(uint32_t x) {
    x ^= x >> 16; x *= 0x7feb352du;
    x ^= x >> 15; x *= 0x846ca68bu;
    x ^= x >> 16; return x;
}

// ---------------------------------------------------------------------------
// Kernel 1: per-(b,t) hypothesis: sample 8 pts, null-space of 9x9 A^T A via
// shifted power iteration, rank-2 projection, normalize, emit F[9] + g=F*diag.
// ---------------------------------------------------------------------------
__global__ void __launch_bounds__(128)
k_estimate(const float* __restrict__ src, const float* __restrict__ dst,
           float* __restrict__ Fg) {
    int id = blockIdx.x * blockDim.x + threadIdx.x;
    if (id >= BATCH * T_PAD) return;
    int b = id / T_PAD, t = id % T_PAD;
    float* out = Fg + (size_t)id * 12;
    if (t >= T_HYP) {                     // padded hypotheses: defined zeros
        #pragma unroll
        for (int i = 0; i < 12; ++i) out[i] = 0.f;
        return;
    }

    // --- deterministic 8 distinct indices ---
    int idx[8];
    uint32_t seed = (uint32_t)b * 0x9E3779B1u ^ (uint32_t)t * 0x85EBCA77u;
    #pragma unroll
    for (int k = 0; k < 8; ++k) {
        uint32_t v = hash32(seed + (uint32_t)k * 0x27D4EB2Fu) & (NPTS - 1);
        for (int tries = 0; tries < 16; ++tries) {
            bool dup = false;
            for (int j = 0; j < k; ++j) dup |= (idx[j] == (int)v);
            if (!dup) break;
            v = hash32(v + 0xB5297A4Du + (uint32_t)tries) & (NPTS - 1);
        }
        idx[k] = (int)v;
    }

    // --- ATA = sum of outer products of the 9-wide design rows ---
    float m[9][9];
    #pragma unroll
    for (int i = 0; i < 9; ++i)
        #pragma unroll
        for (int j = 0; j < 9; ++j) m[i][j] = 0.f;
    for (int r = 0; r < 8; ++r) {
        int p = idx[r];
        float x1 = src[((size_t)b * NPTS + p) * 2 + 0];
        float y1 = src[((size_t)b * NPTS + p) * 2 + 1];
        float x2 = dst[((size_t)b * NPTS + p) * 2 + 0];
        float y2 = dst[((size_t)b * NPTS + p) * 2 + 1];
        float a[9] = {x2*x1, x2*y1, x2, y2*x1, y2*y1, y2, x1, y1, 1.f};
        #pragma unroll
        for (int i = 0; i < 9; ++i)
            #pragma unroll
            for (int j = 0; j < 9; ++j) m[i][j] += a[i] * a[j];
    }
    float tr = 0.f;
    #pragma unroll
    for (int i = 0; i < 9; ++i) tr += m[i][i];

    // --- smallest eigvec of ATA == dominant eigvec of (tr*I - ATA) ---
    float v[9];
    #pragma unroll
    for (int i = 0; i < 9; ++i) v[i] = 1.f + 0.03125f * (float)i;
    for (int it = 0; it < 32; ++it) {
        float w[9];
        #pragma unroll
        for (int i = 0; i < 9; ++i) {
            float s = tr * v[i];
            #pragma unroll
            for (int j = 0; j < 9; ++j) s -= m[i][j] * v[j];
            w[i] = s;
        }
        float n2 = 1e-30f;
        #pragma unroll
        for (int i = 0; i < 9; ++i) n2 += w[i] * w[i];
        float inv = rsqrtf(n2);
        #pragma unroll
        for (int i = 0; i < 9; ++i) v[i] = w[i] * inv;
    }
    float F[9];
    #pragma unroll
    for (int i = 0; i < 9; ++i) F[i] = v[i];

    // --- rank-2: F <- (I - u u^T) F with u = smallest eigvec of F F^T ---
    float M3[3][3];
    #pragma unroll
    for (int i = 0; i < 3; ++i)
        #pragma unroll
        for (int j = 0; j < 3; ++j) {
            float s = 0.f;
            #pragma unroll
            for (int k = 0; k < 3; ++k) s += F[i*3+k] * F[j*3+k];
            M3[i][j] = s;
        }
    float tr3 = M3[0][0] + M3[1][1] + M3[2][2];
    float u0 = 1.f, u1 = 0.75f, u2 = 0.5f;
    for (int it = 0; it < 16; ++it) {
        float w0 = tr3*u0 - (M3[0][0]*u0 + M3[0][1]*u1 + M3[0][2]*u2);
        float w1 = tr3*u1 - (M3[1][0]*u0 + M3[1][1]*u1 + M3[1][2]*u2);
        float w2 = tr3*u2 - (M3[2][0]*u0 + M3[2][1]*u1 + M3[2][2]*u2);
        float inv = rsqrtf(w0*w0 + w1*w1 + w2*w2 + 1e-30f);
        u0 = w0*inv; u1 = w1*inv; u2 = w2*inv;
    }
    float uF0 = u0*F[0] + u1*F[3] + u2*F[6];
    float uF1 = u0*F[1] + u1*F[4] + u2*F[7];
    float uF2 = u0*F[2] + u1*F[5] + u2*F[8];
    F[0] -= u0*uF0; F[1] -= u0*uF1; F[2] -= u0*uF2;
    F[3] -= u1*uF0; F[4] -= u1*uF1; F[5] -= u1*uF2;
    F[6] -= u2*uF0; F[7] -= u2*uF1; F[8] -= u2*uF2;

    float f22 = F[8];
    if (f22 != 0.f) {
        float inv = 1.f / f22;
        #pragma unroll
        for (int i = 0; i < 9; ++i) F[i] *= inv;
    }

    // g = F @ diagvec, diagvec = [src[b,0,0], src[b,1,1], 1] (faithful bug)
    float d0 = src[((size_t)b * NPTS + 0) * 2 + 0];
    float d1 = src[((size_t)b * NPTS + 1) * 2 + 1];
    #pragma unroll
    for (int i = 0; i < 9; ++i) out[i] = F[i];
    out[9]  = F[0]*d0 + F[1]*d1 + F[2];
    out[10] = F[3]*d0 + F[4]*d1 + F[5];
    out[11] = F[6]*d0 + F[7]*d1 + F[8];
}

// ---------------------------------------------------------------------------
// Kernel 2: WMMA Sampson scoring. One wave32 per (batch, 16-hypothesis tile).
// Per 16x16 (point,hyp) tile: 5x V_WMMA_F32_16X16X4_F32 (K=3 padded to 4),
// VALU epilogue counts inliers into the f32 accumulator.
// ---------------------------------------------------------------------------
__global__ void __launch_bounds__(32)
k_score(const float* __restrict__ src, const float* __restrict__ dst,
        const float* __restrict__ Fg, float* __restrict__ counts) {
    const int wave = blockIdx.x;              // 0 .. BATCH*HT-1
    const int b = wave / HT, ht = wave % HT;
    const int lane = threadIdx.x;             // 0..31
    const int n = lane & 15;                  // hyp within tile
    const int hi = ht * 16 + n;
    const bool hiHalf = (lane >= 16);         // A: K=2..3 half, B: K=2..3 half

    const float* fp = Fg + ((size_t)b * T_PAD + hi) * 12;
    float F0 = fp[0], F1 = fp[1], F2 = fp[2];
    float F3 = fp[3], F4 = fp[4], F5 = fp[5];
    float F6 = fp[6], F7 = fp[7];
    float g0 = fp[9], g1 = fp[10], g2 = fp[11];

    // B operands (4x16, K=3 zero-padded): {K0,K1} on lanes 0-15, {K2,0} on 16-31
    v2f Bg, Br0, Br1, Bc0, Bc1;
    Bg.x  = hiHalf ? g2 : g0;  Bg.y  = hiHalf ? 0.f : g1;   // F @ diagvec
    Br0.x = hiHalf ? F2 : F0;  Br0.y = hiHalf ? 0.f : F1;   // F row 0
    Br1.x = hiHalf ? F5 : F3;  Br1.y = hiHalf ? 0.f : F4;   // F row 1
    Bc0.x = hiHalf ? F6 : F0;  Bc0.y = hiHalf ? 0.f : F3;   // F col 0
    Bc1.x = hiHalf ? F7 : F1;  Bc1.y = hiHalf ? 0.f : F4;   // F col 1

    const float* sb = src + (size_t)b * NPTS * 2;
    const float* db = dst + (size_t)b * NPTS * 2;

    const v8f zero = {0.f,0.f,0.f,0.f,0.f,0.f,0.f,0.f};
    v8f cnt = zero;

    for (int pt = 0; pt < NPTS / 16; ++pt) {
        if (pt + 1 < NPTS / 16) {   // global_prefetch_b8 next point tile
            __builtin_prefetch(sb + (pt + 1) * 32, 0, 1);
            __builtin_prefetch(db + (pt + 1) * 32, 0, 1);
        }
        int p = pt * 16 + n;        // point row (both lane halves load same pt)
        float sx = sb[p * 2 + 0], sy = sb[p * 2 + 1];
        float dx = db[p * 2 + 0], dy = db[p * 2 + 1];
        // A operands (16x4): lanes 0-15 hold {x,y} (K0,K1), 16-31 hold {1,0}
        v2f As, Ad;
        As.x = hiHalf ? 1.f : sx;  As.y = hiHalf ? 0.f : sy;
        Ad.x = hiHalf ? 1.f : dx;  Ad.y = hiHalf ? 0.f : dy;

        v8f num = __builtin_amdgcn_wmma_f32_16x16x4_f32(false, Ad, false, Bg,  (short)0, zero, false, false);
        v8f fr0 = __builtin_amdgcn_wmma_f32_16x16x4_f32(false, As, false, Br0, (short)0, zero, false, false);
        v8f fr1 = __builtin_amdgcn_wmma_f32_16x16x4_f32(false, As, false, Br1, (short)0, zero, false, false);
        v8f xc0 = __builtin_amdgcn_wmma_f32_16x16x4_f32(false, Ad, false, Bc0, (short)0, zero, false, false);
        v8f xc1 = __builtin_amdgcn_wmma_f32_16x16x4_f32(false, Ad, false, Bc1, (short)0, zero, false, false);

        #pragma unroll
        for (int e = 0; e < 8; ++e) {
            float den = fr0[e]*fr0[e] + fr1[e]*fr1[e]
                      + xc0[e]*xc0[e] + xc1[e]*xc1[e];
            den = fmaxf(den, 1e-8f);
            float nu = num[e] * num[e];
            cnt[e] += (nu < THRESH2 * den) ? 1.f : 0.f;
        }
    }
    float tot = 0.f;
    #pragma unroll
    for (int e = 0; e < 8; ++e) tot += cnt[e];   // rows e (+8 on hi half)
    tot += __shfl_xor(tot, 16, 32);              // combine the two row halves
    if (lane < 16) counts[(size_t)b * T_PAD + hi] = tot;
}

// ---------------------------------------------------------------------------
// Kernel 3: per batch argmax (first-max tie-break), write best F + mask.
// ---------------------------------------------------------------------------
__global__ void __launch_bounds__(256)
k_select(const float* __restrict__ src, const float* __restrict__ dst,
         const float* __restrict__ Fg, const float* __restrict__ counts,
         float* __restrict__ out) {
    __shared__ int skey[256];
    __shared__ float sF[9];
    __shared__ int sHas;
    const int b = blockIdx.x, tid = threadIdx.x;

    int best = 0;
    for (int t = tid; t < T_HYP; t += 256) {
        int c = (int)counts[(size_t)b * T_PAD + t];
        int key = (c << 11) | (2047 - t);        // max count, then min t
        if (key > best) best = key;
    }
    skey[tid] = best;
    __syncthreads();
    for (int s = 128; s > 0; s >>= 1) {
        if (tid < s && skey[tid + s] > skey[tid]) skey[tid] = skey[tid + s];
        __syncthreads();
    }
    if (tid == 0) {
        int key = skey[0];
        int bt = 2047 - (key & 2047);
        int bc = key >> 11;
        sHas = (bc > 0) ? 1 : 0;
        const float* fp = Fg + ((size_t)b * T_PAD + bt) * 12;
        #pragma unroll
        for (int i = 0; i < 9; ++i) {
            float fv = fp[i];
            sF[i] = fv;
            out[b * 9 + i] = sHas ? fv : 0.f;   // zero F if no inliers
        }
    }
    __syncthreads();

    const float F0 = sF[0], F1 = sF[1], F2 = sF[2];
    const float F3 = sF[3], F4 = sF[4], F5 = sF[5];
    const float F6 = sF[6], F7 = sF[7], F8 = sF[8];
    const int has = sHas;
    const float d0 = src[((size_t)b * NPTS + 0) * 2 + 0];
    const float d1 = src[((size_t)b * NPTS + 1) * 2 + 1];
    const float Fd0 = F0*d0 + F1*d1 + F2;
    const float Fd1 = F3*d0 + F4*d1 + F5;
    const float Fd2 = F6*d0 + F7*d1 + F8;

    for (int p = tid; p < NPTS; p += 256) {
        float sx = src[((size_t)b * NPTS + p) * 2 + 0];
        float sy = src[((size_t)b * NPTS + p) * 2 + 1];
        float dx = dst[((size_t)b * NPTS + p) * 2 + 0];
        float dy = dst[((size_t)b * NPTS + p) * 2 + 1];
        float nu = dx*Fd0 + dy*Fd1 + Fd2;  nu *= nu;
        float fx0 = F0*sx + F1*sy + F2;
        float fx1 = F3*sx + F4*sy + F5;
        float xf0 = dx*F0 + dy*F3 + F6;
        float xf1 = dx*F1 + dy*F4 + F7;
        float den = fmaxf(fx0*fx0 + fx1*fx1 + xf0*xf0 + xf1*xf1, 1e-8f);
        bool inl = (nu < THRESH2 * den) && (has != 0);
        out[72 + (size_t)b * NPTS + p] = inl ? 1.f : 0.f;
    }
}

extern "C" void kernel_launch(void* const* d_in, const int* in_sizes, int n_in,
                              void* d_out, int out_size, void* d_ws, size_t ws_size,
                              hipStream_t stream) {
    (void)in_sizes; (void)n_in; (void)out_size; (void)ws_size;
    const float* src = (const float*)d_in[0];   // [8,1024,2] f32
    const float* dst = (const float*)d_in[1];   // [8,1024,2] f32
    float* Fg     = (float*)d_ws;                               // [8,1008,12]
    float* counts = Fg + (size_t)BATCH * T_PAD * 12;            // [8,1008]
    float* out    = (float*)d_out;              // 72 F floats + 8192 mask floats

    const int n1 = BATCH * T_PAD;               // 8064
    k_estimate<<<(n1 + 127) / 128, 128, 0, stream>>>(src, dst, Fg);
    k_score<<<BATCH * HT, 32, 0, stream>>>(src, dst, Fg, counts);
    k_select<<<BATCH, 256, 0, stream>>>(src, dst, Fg, counts, out);
}